// DeepComplexRBF_82248623718501
// MI455X (gfx1250) — compile-verified
//
#include <hip/hip_runtime.h>
#include <hip/hip_bf16.h>
#include <math.h>

// ---------------------------------------------------------------------------
// DeepComplexRBF on MI455X (gfx1250): bandwidth-bound GEMV pipeline.
//   dist_sq[i] = ||y||^2 + ||G_i||^2 - 2*(Gr[i,:]·yr + Gi[i,:]·yi)
//   phi = exp(-dist_sq/(2s));  y' = W@phi + b   (complex W, real phi)
// All matrices streamed once (~576 MB) -> ~25us at 23.3 TB/s.
// Dot engine: v_wmma_f32_16x16x4_f32, 16 rows per wave, split-K partials
// (deterministic, no float atomics). Row sum-of-squares rides along on VALU.
// ---------------------------------------------------------------------------

typedef __attribute__((ext_vector_type(2))) float v2f;
typedef __attribute__((ext_vector_type(8))) float v8f;

#define CKSZ   256            // K elements per split-K chunk
#define MAXI   4096
#define MAXKC  16             // 4096 / 256

// Wave-level GEMV core over k in [kbeg,kend) for 16 rows.
// A VGPR layout (f32 16x16x4): lanes 0-15 -> (M=lane, K=0,1), lanes 16-31 ->
// (M=lane-16, K=2,3). B replicated across all N columns => same addressing
// with the row term dropped. C accumulates across iterations.
template <bool WITHSQ>
__device__ __forceinline__ void gemv_core(const float* __restrict__ mrow,  // row base + koff
                                          const float* __restrict__ vk,   // vec + koff
                                          int kbeg, int kend,
                                          v8f& c, float& sqp)
{
#pragma unroll 4
    for (int k = kbeg; k < kend; k += 4) {
        float2 a = *reinterpret_cast<const float2*>(mrow + k);
        float2 b = *reinterpret_cast<const float2*>(vk + k);
        v2f av; av[0] = a.x; av[1] = a.y;
        v2f bv; bv[0] = b.x; bv[1] = b.y;
        // D = A(16x4) * B(4x16) + C ; emits v_wmma_f32_16x16x4_f32
        c = __builtin_amdgcn_wmma_f32_16x16x4_f32(false, av, false, bv,
                                                  (short)0, c, false, false);
        if (WITHSQ) {
            sqp = fmaf(a.x, a.x, sqp);
            sqp = fmaf(a.y, a.y, sqp);
        }
    }
}

// Phase A: for 16 rows per wave, accumulate over both (real,imag) planes of G:
//   dotP[kc][i] = sum_k Gr[i,k]*yr[k] + Gi[i,k]*yi[k]   (partial over k-chunk)
//   sqP [kc][i] = sum_k Gr[i,k]^2 + Gi[i,k]^2
__global__ void __launch_bounds__(256)
phaseA_kernel(const float* __restrict__ G, const float* __restrict__ y,
              float* __restrict__ dotP, float* __restrict__ sqP,
              int K, int I)
{
    const int wave = threadIdx.x >> 5;
    const int lane = threadIdx.x & 31;
    const int row0 = blockIdx.x * 128 + wave * 16;
    const int kc   = blockIdx.y;
    const int kbeg = kc * CKSZ;
    const int kend = kbeg + CKSZ;
    const int r    = lane & 15;
    const int koff = (lane >> 4) << 1;     // 0 or 2

    v8f c; for (int j = 0; j < 8; ++j) c[j] = 0.0f;
    float sqp = 0.0f;

#pragma unroll
    for (int part = 0; part < 2; ++part) {
        const float* mrow = G + (size_t)part * I * K + (size_t)(row0 + r) * K + koff;
        const float* vk   = y + (size_t)part * K + koff;
        gemv_core<true>(mrow, vk, kbeg, kend, c, sqp);
    }

    // row sumsq: lane L (<16) has k%4 in {0,1}, lane L+16 has {2,3} of row row0+L
    float osq = __shfl_xor(sqp, 16, 32);
    if (lane < 16) sqP[(size_t)kc * I + row0 + lane] = sqp + osq;

    // D col 0: lane 0 -> rows M=0..7 in c[0..7]; lane 16 -> rows M=8..15
    if ((lane & 15) == 0) {
        const int mbase = row0 + ((lane >> 4) << 3);
        float* dp = dotP + (size_t)kc * I + mbase;
#pragma unroll
        for (int j = 0; j < 8; ++j) dp[j] = c[j];
    }
}

// Phase B: yaccP[part][kc][o] = sum_k W[part][o,k] * phi[k]  (partial)
__global__ void __launch_bounds__(256)
phaseB_kernel(const float* __restrict__ W, const float* __restrict__ phi,
              float* __restrict__ yaccP, int K, int O, int KC)
{
    const int wave = threadIdx.x >> 5;
    const int lane = threadIdx.x & 31;
    const int row0 = blockIdx.x * 128 + wave * 16;
    const int kc   = blockIdx.y;
    const int part = blockIdx.z;
    const int kbeg = kc * CKSZ;
    const int kend = kbeg + CKSZ;
    const int r    = lane & 15;
    const int koff = (lane >> 4) << 1;

    v8f c; for (int j = 0; j < 8; ++j) c[j] = 0.0f;
    float sqp = 0.0f;

    const float* mrow = W + (size_t)part * O * K + (size_t)(row0 + r) * K + koff;
    const float* vk   = phi + koff;
    gemv_core<false>(mrow, vk, kbeg, kend, c, sqp);

    if ((lane & 15) == 0) {
        const int mbase = row0 + ((lane >> 4) << 3);
        float* dp = yaccP + ((size_t)part * KC + kc) * O + mbase;
#pragma unroll
        for (int j = 0; j < 8; ++j) dp[j] = c[j];
    }
}

// phi[i] = exp(-( ||y||^2 + sq[i] - 2*dot[i] ) / (2*s[i])), reducing split-K partials.
__global__ void __launch_bounds__(256)
phi_kernel(const float* __restrict__ dotP, const float* __restrict__ sqP,
           const float* __restrict__ s, const float* __restrict__ normP,
           int nnorm, float* __restrict__ phi, int I, int KC)
{
    const int i = blockIdx.x * 256 + threadIdx.x;
    if (i >= I) return;
    float yn = 0.0f;
    for (int j = 0; j < nnorm; ++j) yn += normP[j];
    float d = 0.0f, q = 0.0f;
    for (int kc = 0; kc < KC; ++kc) {
        d += dotP[(size_t)kc * I + i];
        q += sqP[(size_t)kc * I + i];
    }
    const float dist = yn + q - 2.0f * d;
    phi[i] = __expf(-dist / (2.0f * s[i]));
}

// y_out = reduce(yaccP) + b ; also emit per-block partial of ||y_out||^2.
__global__ void __launch_bounds__(256)
bias_kernel(const float* __restrict__ yaccP, const float* __restrict__ b,
            float* __restrict__ yout, float* __restrict__ normP, int O, int KC)
{
    __shared__ float red[256];
    const int o = blockIdx.x * 256 + threadIdx.x;   // O is a multiple of 256
    float yr = b[o], yi = b[O + o];
    for (int kc = 0; kc < KC; ++kc) {
        yr += yaccP[(size_t)kc * O + o];
        yi += yaccP[((size_t)KC + kc) * O + o];
    }
    yout[o]     = yr;
    yout[O + o] = yi;

    red[threadIdx.x] = yr * yr + yi * yi;
    __syncthreads();
    for (int st = 128; st > 0; st >>= 1) {
        if (threadIdx.x < st) red[threadIdx.x] += red[threadIdx.x + st];
        __syncthreads();
    }
    if (threadIdx.x == 0) normP[blockIdx.x] = red[0];
}

// ||x||^2 of the (2, N) input vector -> normP[0].
__global__ void __launch_bounds__(256)
norm0_kernel(const float* __restrict__ x, int n, float* __restrict__ normP)
{
    __shared__ float red[256];
    float a = 0.0f;
    for (int i = threadIdx.x; i < n; i += 256) { float v = x[i]; a = fmaf(v, v, a); }
    red[threadIdx.x] = a;
    __syncthreads();
    for (int st = 128; st > 0; st >>= 1) {
        if (threadIdx.x < st) red[threadIdx.x] += red[threadIdx.x + st];
        __syncthreads();
    }
    if (threadIdx.x == 0) normP[0] = red[0];
}

extern "C" void kernel_launch(void* const* d_in, const int* in_sizes, int n_in,
                              void* d_out, int out_size, void* d_ws, size_t ws_size,
                              hipStream_t stream)
{
    (void)in_sizes; (void)n_in; (void)out_size; (void)ws_size;

    const float* x = (const float*)d_in[0];
    const float* W[3] = { (const float*)d_in[1], (const float*)d_in[5], (const float*)d_in[9]  };
    const float* B[3] = { (const float*)d_in[2], (const float*)d_in[6], (const float*)d_in[10] };
    const float* G[3] = { (const float*)d_in[3], (const float*)d_in[7], (const float*)d_in[11] };
    const float* S[3] = { (const float*)d_in[4], (const float*)d_in[8], (const float*)d_in[12] };

    const int Oprev[3] = { 1024, 4096, 4096 };   // K of phase A (input dim of layer)
    const int Il[3]    = { 4096, 4096, 4096 };   // rows of G / K of phase B
    const int Ol[3]    = { 4096, 4096, 1024 };   // rows of W

    // workspace layout (floats)
    float* ws    = (float*)d_ws;
    float* dotP  = ws;                   // [MAXKC][MAXI]   = 65536
    float* sqP   = ws + 65536;           // [MAXKC][MAXI]   = 65536
    float* phi   = ws + 131072;          // [MAXI]          = 4096
    float* yaccP = ws + 135168;          // [2][MAXKC][MAXI]= 131072
    float* ybufA = ws + 266240;          // (2,4096)        = 8192
    float* ybufB = ws + 274432;          // (2,4096)        = 8192
    float* normP = ws + 282624;          // up to 16 block partials
    float* out   = (float*)d_out;        // (2, 1024)

    norm0_kernel<<<1, 256, 0, stream>>>(x, 2 * 1024, normP);
    int nnorm = 1;
    const float* yin = x;

    for (int l = 0; l < 3; ++l) {
        const int K = Oprev[l], I = Il[l], O = Ol[l];

        const int KCa = K / CKSZ;
        dim3 gA((unsigned)(I / 128), (unsigned)KCa);
        phaseA_kernel<<<gA, 256, 0, stream>>>(G[l], yin, dotP, sqP, K, I);

        phi_kernel<<<(unsigned)(I / 256), 256, 0, stream>>>(dotP, sqP, S[l], normP,
                                                            nnorm, phi, I, KCa);

        const int KCb = I / CKSZ;
        dim3 gB((unsigned)(O / 128), (unsigned)KCb, 2);
        phaseB_kernel<<<gB, 256, 0, stream>>>(W[l], phi, yaccP, I, O, KCb);

        float* yout = (l == 0) ? ybufA : (l == 1) ? ybufB : out;
        bias_kernel<<<(unsigned)(O / 256), 256, 0, stream>>>(yaccP, B[l], yout, normP, O, KCb);

        nnorm = O / 256;
        yin = yout;
    }
}